// PFNLayerV2_60000693125782
// MI455X (gfx1250) — compile-verified
//
#include <hip/hip_runtime.h>
#include <hip/hip_bf16.h>

typedef float v2f __attribute__((ext_vector_type(2)));
typedef float v8f __attribute__((ext_vector_type(8)));

#define BN_EPS   1e-3f
#define IN_CH    64
#define HID_CH   64
#define NUM_SEG  40000

// ---------------------------------------------------------------- zero seg
__global__ void k_zero_f4(float4* __restrict__ p, int n4) {
  int i = blockIdx.x * blockDim.x + threadIdx.x;
  if (i < n4) p[i] = make_float4(0.f, 0.f, 0.f, 0.f);
}

// ---------------------------------------------------------------- GEMM + batch stats
// One wave per 16-row tile. 8 waves / block. x (raw) -> out[:, 0:64] (stride 128).
// Per-block channel sums/sumsq -> partials[block*128 + {0..63 sum, 64..127 sumsq}].
__global__ __launch_bounds__(256) void k_gemm(const float* __restrict__ A,
                                              const float* __restrict__ W,
                                              float* __restrict__ out,
                                              float* __restrict__ partials,
                                              int nRows) {
  __shared__ float ldsW[64 * 64];        // W row-major [k][n], 16KB
  __shared__ float ldsA[8][16 * 64];     // per-wave A tile, later reused as C tile
  __shared__ float sSum[64];
  __shared__ float sSq[64];

  const int tid  = threadIdx.x;
  const int wave = tid >> 5;
  const int lane = tid & 31;

  if (tid < 64) { sSum[tid] = 0.f; sSq[tid] = 0.f; }

  // Load W (4096 floats) cooperatively, coalesced float4.
  for (int it = 0; it < 4; ++it) {
    int idx = (it * 256 + tid) * 4;
    *(float4*)&ldsW[idx] = *(const float4*)&W[idx];
  }
  __syncthreads();

  const int  tile     = blockIdx.x * 8 + wave;
  const int  row0     = tile * 16;
  const bool active   = row0 < nRows;            // wave-uniform
  const bool fullTile = (row0 + 16) <= nRows;    // wave-uniform

  if (active) {
    // Stage 16x64 A tile, coalesced (2 rows per 128B-lane issue).
    if (fullTile) {
#pragma unroll
      for (int it = 0; it < 8; ++it) {
        int idx = it * 32 + lane;                // 0..255
        int r   = idx >> 4;
        int c   = (idx & 15) * 4;
        *(float4*)&ldsA[wave][r * 64 + c] = *(const float4*)&A[(size_t)(row0 + r) * 64 + c];
      }
    } else {
      for (int it = 0; it < 8; ++it) {
        int idx = it * 32 + lane;
        int r   = idx >> 4;
        int c   = (idx & 15) * 4;
        if (row0 + r < nRows)
          *(float4*)&ldsA[wave][r * 64 + c] = *(const float4*)&A[(size_t)(row0 + r) * 64 + c];
        else
          *(float4*)&ldsA[wave][r * 64 + c] = make_float4(0.f, 0.f, 0.f, 0.f);
      }
    }
  }
  __syncthreads();   // cross-lane LDS visibility (all waves reach this)

  if (active) {
    const float* aT = ldsA[wave];
    const int m  = lane & 15;
    const int kh = (lane >> 4) * 2;          // lanes 16-31 hold K+2,K+3

    // A fragments for all 16 K-steps: lane m holds A[m][k], A[m][k+1].
    v2f afrag[16];
#pragma unroll
    for (int kt = 0; kt < 16; ++kt)
      afrag[kt] = *(const v2f*)&aT[m * 64 + kt * 4 + kh];

    const int nSub = lane & 15;
#pragma unroll
    for (int ct = 0; ct < 4; ++ct) {
      const int n = ct * 16 + nSub;
      v8f acc = {};
#pragma unroll
      for (int kt = 0; kt < 16; ++kt) {
        const int k0 = kt * 4 + kh;
        v2f bfrag;                           // B[k][n], B[k+1][n]
        bfrag.x = ldsW[(k0 + 0) * 64 + n];
        bfrag.y = ldsW[(k0 + 1) * 64 + n];
        acc = __builtin_amdgcn_wmma_f32_16x16x4_f32(
            false, afrag[kt], false, bfrag, (short)0, acc, false, false);
      }
      // Spill C tile to LDS (reuse A region) + accumulate stats.
      // VGPR v, lanes 0-15 -> M=v; lanes 16-31 -> M=v+8. N = n.
      float s = 0.f, sq = 0.f;
#pragma unroll
      for (int v = 0; v < 8; ++v) {
        const int mm = v + (lane >> 4) * 8;
        float x = acc[v];
        ldsA[wave][mm * 64 + n] = x;
        s  += x;
        sq += x * x;
      }
      // lane l and l+16 share channel n: fold, then LDS-atomic per channel.
      s  += __shfl_xor(s, 16, 32);
      sq += __shfl_xor(sq, 16, 32);
      if (lane < 16) {
        atomicAdd(&sSum[n], s);
        atomicAdd(&sSq[n],  sq);
      }
    }
  }
  __syncthreads();   // C tile visible across lanes; stats atomics done

  if (active) {
    // Coalesced writeback: 16 rows x 64 ch, 256B contiguous per row, b128 stores.
    if (fullTile) {
#pragma unroll
      for (int it = 0; it < 8; ++it) {
        int idx = it * 32 + lane;                // 0..255
        int r   = idx >> 4;
        int c   = (idx & 15) * 4;
        *(float4*)&out[(size_t)(row0 + r) * 128 + c] = *(const float4*)&ldsA[wave][r * 64 + c];
      }
    } else {
      for (int it = 0; it < 8; ++it) {
        int idx = it * 32 + lane;
        int r   = idx >> 4;
        int c   = (idx & 15) * 4;
        if (row0 + r < nRows)
          *(float4*)&out[(size_t)(row0 + r) * 128 + c] = *(const float4*)&ldsA[wave][r * 64 + c];
      }
    }
  }

  if (tid < 64) {
    partials[(size_t)blockIdx.x * 128 + tid]      = sSum[tid];
    partials[(size_t)blockIdx.x * 128 + 64 + tid] = sSq[tid];
  }
}

// ---------------------------------------------------------------- stats reduce -> scale/bias
__global__ __launch_bounds__(1024) void k_stats(const float* __restrict__ partials, int nBlocks,
                                                const float* __restrict__ gamma,
                                                const float* __restrict__ beta,
                                                float* __restrict__ scaleBias, float invN) {
  __shared__ float rs[1024];
  __shared__ float rq[1024];
  const int t     = threadIdx.x;
  const int c     = t & 63;
  const int chunk = t >> 6;                  // 16 parallel chunks per channel
  float s = 0.f, sq = 0.f;
  for (int b = chunk; b < nBlocks; b += 16) {
    s  += partials[(size_t)b * 128 + c];
    sq += partials[(size_t)b * 128 + 64 + c];
  }
  rs[t] = s; rq[t] = sq;
  __syncthreads();
  for (int off = 512; off >= 64; off >>= 1) {
    if (t < off) { rs[t] += rs[t + off]; rq[t] += rq[t + off]; }
    __syncthreads();
  }
  if (t < 64) {
    float mean = rs[t] * invN;
    float var  = rq[t] * invN - mean * mean;   // biased var (matches jnp.var)
    float sc   = gamma[t] * rsqrtf(var + BN_EPS);
    scaleBias[t]      = sc;
    scaleBias[64 + t] = beta[t] - mean * sc;
  }
}

// ---------------------------------------------------------------- BN + ReLU + segment max
// 16 threads x float4 cover 64 channels; each thread walks 16 consecutive points,
// merging same-segment maxima locally before flushing one atomicMax (indices sorted).
__global__ __launch_bounds__(256) void k_bn_seg(float* __restrict__ out,
                                                const long long* __restrict__ inv,
                                                const float* __restrict__ scaleBias,
                                                float* __restrict__ seg, int nRows) {
  const int tid   = threadIdx.x;
  const int c0    = (tid & 15) * 4;
  const int group = blockIdx.x * 16 + (tid >> 4);
  const int pbase = group * 16;
  if (pbase >= nRows) return;

  const float4 sc = *(const float4*)&scaleBias[c0];
  const float4 bs = *(const float4*)&scaleBias[64 + c0];
  unsigned int* segU = (unsigned int*)seg;

  float4 run = make_float4(0.f, 0.f, 0.f, 0.f);
  long long cur = -1;
  for (int j = 0; j < 16; ++j) {
    const int i = pbase + j;
    if (i >= nRows) break;
    float4 x = *(float4*)&out[(size_t)i * 128 + c0];
    x.x = fmaxf(fmaf(x.x, sc.x, bs.x), 0.f);
    x.y = fmaxf(fmaf(x.y, sc.y, bs.y), 0.f);
    x.z = fmaxf(fmaf(x.z, sc.z, bs.z), 0.f);
    x.w = fmaxf(fmaf(x.w, sc.w, bs.w), 0.f);
    *(float4*)&out[(size_t)i * 128 + c0] = x;

    const long long s = inv[i];
    if (s != cur) {
      if (cur >= 0) {
        const size_t b = (size_t)cur * 64 + c0;
        atomicMax(&segU[b + 0], __float_as_uint(run.x));
        atomicMax(&segU[b + 1], __float_as_uint(run.y));
        atomicMax(&segU[b + 2], __float_as_uint(run.z));
        atomicMax(&segU[b + 3], __float_as_uint(run.w));
      }
      cur = s;
      run = make_float4(0.f, 0.f, 0.f, 0.f);
    }
    run.x = fmaxf(run.x, x.x);
    run.y = fmaxf(run.y, x.y);
    run.z = fmaxf(run.z, x.z);
    run.w = fmaxf(run.w, x.w);
  }
  if (cur >= 0) {
    const size_t b = (size_t)cur * 64 + c0;
    atomicMax(&segU[b + 0], __float_as_uint(run.x));
    atomicMax(&segU[b + 1], __float_as_uint(run.y));
    atomicMax(&segU[b + 2], __float_as_uint(run.z));
    atomicMax(&segU[b + 3], __float_as_uint(run.w));
  }
}

// ---------------------------------------------------------------- gather voxel max back
__global__ __launch_bounds__(256) void k_gather(float* __restrict__ out,
                                                const long long* __restrict__ inv,
                                                const float* __restrict__ seg, int nRows) {
  const int tid = threadIdx.x;
  const int c0  = (tid & 15) * 4;
  const int i   = blockIdx.x * 16 + (tid >> 4);
  if (i >= nRows) return;
  const long long s = inv[i];
  const float4 v = *(const float4*)&seg[(size_t)s * 64 + c0];
  *(float4*)&out[(size_t)i * 128 + 64 + c0] = v;
}

// ---------------------------------------------------------------- launcher
extern "C" void kernel_launch(void* const* d_in, const int* in_sizes, int n_in,
                              void* d_out, int out_size, void* d_ws, size_t ws_size,
                              hipStream_t stream) {
  const float*     inputs = (const float*)d_in[0];
  const float*     W      = (const float*)d_in[1];
  const float*     gamma  = (const float*)d_in[2];
  const float*     beta   = (const float*)d_in[3];
  const long long* inv    = (const long long*)d_in[4];
  float* out = (float*)d_out;

  const int nRows    = in_sizes[0] / IN_CH;       // 1,000,000
  const int nTiles   = (nRows + 15) / 16;
  const int nBlocks1 = (nTiles + 7) / 8;

  char*  ws        = (char*)d_ws;
  float* seg       = (float*)ws;                                  // NUM_SEG*64 f32 (10.24 MB)
  float* partials  = (float*)(ws + (size_t)NUM_SEG * 64 * 4);     // nBlocks1*128 f32
  float* scaleBias = partials + (size_t)nBlocks1 * 128;           // 128 f32

  const int segN4 = NUM_SEG * 64 / 4;
  k_zero_f4<<<(segN4 + 255) / 256, 256, 0, stream>>>((float4*)seg, segN4);

  k_gemm<<<nBlocks1, 256, 0, stream>>>(inputs, W, out, partials, nRows);

  k_stats<<<1, 1024, 0, stream>>>(partials, nBlocks1, gamma, beta, scaleBias,
                                  1.0f / (float)nRows);

  const int nGroups = (nRows + 15) / 16;
  k_bn_seg<<<(nGroups + 15) / 16, 256, 0, stream>>>(out, inv, scaleBias, seg, nRows);

  k_gather<<<(nRows + 15) / 16, 256, 0, stream>>>(out, inv, seg, nRows);
}